// _OrderSimilarity_15899968930480
// MI455X (gfx1250) — compile-verified
//
#include <hip/hip_runtime.h>
#include <stdint.h>

// ---------------------------------------------------------------------------
// order-similarity: out[i,j] = -|| relu(lab[j,:] - vis[i,:]) ||_2
//   lab: (M=512, D=512), vis: (N=1024, D=512), out: (N, M)
//
// Compute-bound elementwise-nonlinear reduction (ReLU inside the sum => not
// expressible as a matmul / WMMA). Optimized as a register-blocked VALU
// kernel with gfx1250 async global->LDS double buffering (ASYNCcnt path).
// ---------------------------------------------------------------------------

#define TILE      64
#define DC        32
#define LDS_PITCH 36   // DC + 4 pad floats; row = 144 B (16B-aligned, bank-staggered)

__device__ __forceinline__ uint32_t lds_off(const void* p) {
  // Low 32 bits of a generic pointer into __shared__ == LDS byte address
  // (addrspacecast 3->0 is {aperture_hi, lds_offset} on gfx1250).
  return (uint32_t)(uintptr_t)p;
}

__device__ __forceinline__ void async_ld_b128(uint32_t lds, const float* gp) {
  uint64_t ga = (uint64_t)(uintptr_t)gp;
  asm volatile("global_load_async_to_lds_b128 %0, %1, off"
               :: "v"(lds), "v"(ga) : "memory");
}

__global__ __launch_bounds__(256) void order_sim_tile64(
    const float* __restrict__ lab,  // (M, D) row-major
    const float* __restrict__ vis,  // (N, D) row-major
    float* __restrict__ out,        // (N, M) row-major
    int M, int D) {
  __shared__ __align__(16) float s_lab[2][TILE][LDS_PITCH];
  __shared__ __align__(16) float s_vis[2][TILE][LDS_PITCH];

  const int tid = threadIdx.x;
  const int tx  = tid & 15;   // -> lab (j) sub-index
  const int ty  = tid >> 4;   // -> vis (i) sub-index
  const int jb  = blockIdx.x * TILE;  // lab-tile origin
  const int ib  = blockIdx.y * TILE;  // vis-tile origin

  // Async-copy ownership: each thread owns one tile row and two 16B column
  // groups -> exactly 4 async b128 copies per chunk per thread (per wave:
  // ASYNCcnt +4 per chunk).
  const int crow = tid >> 2;        // 0..63
  const int g0   = (tid & 3) * 4;   // 0,4,8,12
  const int g1   = g0 + 16;         // 16,20,24,28

  const float* glab = lab + (size_t)(jb + crow) * D;
  const float* gvis = vis + (size_t)(ib + crow) * D;

  // Prologue: chunk 0 -> buffer 0.
  async_ld_b128(lds_off(&s_lab[0][crow][g0]), glab + g0);
  async_ld_b128(lds_off(&s_lab[0][crow][g1]), glab + g1);
  async_ld_b128(lds_off(&s_vis[0][crow][g0]), gvis + g0);
  async_ld_b128(lds_off(&s_vis[0][crow][g1]), gvis + g1);

  float acc[4][4];
#pragma unroll
  for (int a = 0; a < 4; ++a)
#pragma unroll
    for (int b = 0; b < 4; ++b) acc[a][b] = 0.0f;

  const int nch = D / DC;   // 16
  for (int k = 0; k < nch; ++k) {
    if (k + 1 < nch) {
      const int nb = (k + 1) & 1;
      const int d0 = (k + 1) * DC;
      async_ld_b128(lds_off(&s_lab[nb][crow][g0]), glab + d0 + g0);
      async_ld_b128(lds_off(&s_lab[nb][crow][g1]), glab + d0 + g1);
      async_ld_b128(lds_off(&s_vis[nb][crow][g0]), gvis + d0 + g0);
      async_ld_b128(lds_off(&s_vis[nb][crow][g1]), gvis + d0 + g1);
      // Async loads retire in order: <=4 outstanding => this wave's chunk-k
      // copies are done while chunk-(k+1)'s four stay in flight.
      asm volatile("s_wait_asynccnt 0x4" ::: "memory");
    } else {
      asm volatile("s_wait_asynccnt 0x0" ::: "memory");
    }
    __syncthreads();  // all waves' chunk-k tiles visible in LDS

    const int buf = k & 1;
#pragma unroll
    for (int d4 = 0; d4 < DC; d4 += 4) {
      float4 a[4], b[4];
#pragma unroll
      for (int r = 0; r < 4; ++r)
        a[r] = *(const float4*)&s_lab[buf][tx + 16 * r][d4];  // ds_load_b128
#pragma unroll
      for (int r = 0; r < 4; ++r)
        b[r] = *(const float4*)&s_vis[buf][ty + 16 * r][d4];  // broadcast read
#pragma unroll
      for (int ri = 0; ri < 4; ++ri) {
#pragma unroll
        for (int rj = 0; rj < 4; ++rj) {
          float t;
          t = a[rj].x - b[ri].x; t = fmaxf(t, 0.0f); acc[ri][rj] = fmaf(t, t, acc[ri][rj]);
          t = a[rj].y - b[ri].y; t = fmaxf(t, 0.0f); acc[ri][rj] = fmaf(t, t, acc[ri][rj]);
          t = a[rj].z - b[ri].z; t = fmaxf(t, 0.0f); acc[ri][rj] = fmaf(t, t, acc[ri][rj]);
          t = a[rj].w - b[ri].w; t = fmaxf(t, 0.0f); acc[ri][rj] = fmaf(t, t, acc[ri][rj]);
        }
      }
    }
    __syncthreads();  // this buffer is re-targeted by chunk k+2's async copies
  }

#pragma unroll
  for (int ri = 0; ri < 4; ++ri) {
    const int i = ib + ty + 16 * ri;
#pragma unroll
    for (int rj = 0; rj < 4; ++rj) {
      const int j = jb + tx + 16 * rj;
      out[(size_t)i * M + j] = -sqrtf(acc[ri][rj]);
    }
  }
}

extern "C" void kernel_launch(void* const* d_in, const int* in_sizes, int n_in,
                              void* d_out, int out_size, void* d_ws, size_t ws_size,
                              hipStream_t stream) {
  const float* lab = (const float*)d_in[0];  // (512, 512)
  const float* vis = (const float*)d_in[1];  // (1024, 512)
  float* out = (float*)d_out;                // (1024, 512)

  const int D = 512;
  const int M = in_sizes[0] / D;  // 512
  const int N = in_sizes[1] / D;  // 1024

  dim3 grid((unsigned)(M / TILE), (unsigned)(N / TILE));  // (8, 16) = 128 blocks
  order_sim_tile64<<<grid, 256, 0, stream>>>(lab, vis, out, M, D);
}